// BaseAttention_23570780520709
// MI455X (gfx1250) — compile-verified
//
#include <hip/hip_runtime.h>
#include <math.h>

// MI455X / gfx1250: wave32, WMMA 16x16x4 fp32 matrix pipe + async global->LDS.
typedef __attribute__((ext_vector_type(2))) float v2f;
typedef __attribute__((ext_vector_type(8))) float v8f;
typedef int v4i __attribute__((__vector_size__(16)));
typedef __attribute__((address_space(1))) v4i* gas_v4i_ptr;  // global (AS1)
typedef __attribute__((address_space(3))) v4i* las_v4i_ptr;  // LDS (AS3)

#define D_MODEL 1024
#define T_SEQ   2048
#define NHEAD   16
#define HDIM    64
#define BATCH   4
#define KTPL    4

#if __has_builtin(__builtin_amdgcn_global_load_async_to_lds_b128) && \
    __has_builtin(__builtin_amdgcn_s_wait_asynccnt)
#define ASYNC_COPY 1
#define WAIT_ASYNC(n) __builtin_amdgcn_s_wait_asynccnt(n)
#else
#define ASYNC_COPY 0
#define WAIT_ASYNC(n)
#endif

__device__ __forceinline__ v8f wmma4(v2f a, v2f b, v8f c) {
  // D = A(16x4,f32) * B(4x16,f32) + C(16x16,f32)
  return __builtin_amdgcn_wmma_f32_16x16x4_f32(false, a, false, b, (short)0, c,
                                               false, false);
}

// 16-byte global -> LDS copy (async on gfx1250, tracked by ASYNCcnt).
__device__ __forceinline__ void cp16(const float* __restrict__ g,
                                     float* __restrict__ l) {
#if ASYNC_COPY
  __builtin_amdgcn_global_load_async_to_lds_b128(
      (gas_v4i_ptr)(void*)g, (las_v4i_ptr)(void*)l, 0, 0);
#else
  *(float4*)l = *(const float4*)g;
#endif
}

// ---------------------------------------------------------------------------
// GEMM: C[M,N] = A[M,K] @ Bw[N,K]^T (+ bias[N]).
// WG = 256 threads (8 waves), tile 64(M) x 256(N), K-chunk 16, double-buffered
// LDS staging via async global->LDS b128 copies (5 per thread per chunk).
// Wave tile = 32(M) x 64(N): per k-step 6 ds_load_b64 feed 8 WMMAs.
// LDS row pitch 20 floats keeps both b128 stores (16B) and b64 reads (8B)
// aligned. 51.2 KB LDS per WG (320 KB per WGP -> ~6 WGs resident).
// ---------------------------------------------------------------------------
__global__ __launch_bounds__(256) void gemm_nt_kernel(
    const float* __restrict__ A, const float* __restrict__ Bw,
    const float* __restrict__ bias, float* __restrict__ C,
    int M, int N, int K) {
  __shared__ __align__(16) float lA[2][64 * 20];   // 64 rows x KC(16) pitch 20
  __shared__ __align__(16) float lB[2][256 * 20];  // 256 rows x KC(16) pitch 20

  const int t = threadIdx.x;
  const int lane = t & 31;
  const int wave = t >> 5;
  const int lo = lane & 15;
  const int hi = lane >> 4;
  const int wm = wave & 1;   // 2 waves along M
  const int wn = wave >> 1;  // 4 waves along N
  const int mBlock = blockIdx.y * 64;
  const int nBlock = blockIdx.x * 256;
  (void)M;

  // copy assignment: thread -> (row, 4-float k-segment)
  const int crow = t >> 2;
  const int ckseg = (t & 3) * 4;

  v8f acc[2][4];
#pragma unroll
  for (int mt = 0; mt < 2; ++mt)
#pragma unroll
    for (int nt = 0; nt < 4; ++nt) acc[mt][nt] = (v8f)(0.0f);

  const int nch = K / 16;

  // issue copies for chunk c into buffer bi (5 x b128 per thread)
  auto issue = [&](int c, int bi) {
    const int kc = c * 16;
    cp16(A + (size_t)(mBlock + crow) * K + kc + ckseg,
         &lA[bi][crow * 20 + ckseg]);
#pragma unroll
    for (int r = 0; r < 4; ++r) {
      const int br = crow + 64 * r;
      cp16(Bw + (size_t)(nBlock + br) * K + kc + ckseg,
           &lB[bi][br * 20 + ckseg]);
    }
  };

  issue(0, 0);

  const int ar0 = (wm * 32 + lo) * 20 + 2 * hi;
  const int ar1 = ar0 + 16 * 20;
  const int br0 = (wn * 64 + lo) * 20 + 2 * hi;

  for (int c = 0; c < nch; ++c) {
    const int buf = c & 1;
    if (c + 1 < nch) {
      issue(c + 1, buf ^ 1);
      WAIT_ASYNC(5);  // in-order completion -> chunk c landed
    } else {
      WAIT_ASYNC(0);
    }
    __syncthreads();

    const float* As = &lA[buf][0];
    const float* Bs = &lB[buf][0];
#pragma unroll
    for (int kk = 0; kk < 16; kk += 4) {
      v2f a0 = *(const v2f*)(As + ar0 + kk);
      v2f a1 = *(const v2f*)(As + ar1 + kk);
      v2f b0 = *(const v2f*)(Bs + br0 + kk);
      v2f b1 = *(const v2f*)(Bs + br0 + 16 * 20 + kk);
      v2f b2 = *(const v2f*)(Bs + br0 + 32 * 20 + kk);
      v2f b3 = *(const v2f*)(Bs + br0 + 48 * 20 + kk);
      acc[0][0] = wmma4(a0, b0, acc[0][0]);
      acc[1][0] = wmma4(a1, b0, acc[1][0]);
      acc[0][1] = wmma4(a0, b1, acc[0][1]);
      acc[1][1] = wmma4(a1, b1, acc[1][1]);
      acc[0][2] = wmma4(a0, b2, acc[0][2]);
      acc[1][2] = wmma4(a1, b2, acc[1][2]);
      acc[0][3] = wmma4(a0, b3, acc[0][3]);
      acc[1][3] = wmma4(a1, b3, acc[1][3]);
    }
    __syncthreads();  // all waves done reading buf before it is overwritten
  }

#pragma unroll
  for (int mt = 0; mt < 2; ++mt)
#pragma unroll
    for (int nt = 0; nt < 4; ++nt)
#pragma unroll
      for (int j = 0; j < 8; ++j) {
        const int row = mBlock + wm * 32 + mt * 16 + j + 8 * hi;
        const int col = nBlock + wn * 64 + nt * 16 + lo;
        float v = acc[mt][nt][j];
        if (bias) v += bias[col];
        C[(size_t)row * N + col] = v;
      }
}

// ---------------------------------------------------------------------------
// Flash attention: one wave owns 16 q-rows of one (b,h).  Block = 4 waves.
// qkv layout: [B,T,3*D] with Q at col h*64, K at 1024+h*64, V at 2048+h*64.
// Per k-tile: batch 16 K-operand loads before the 16 chained QK WMMAs, and
// batch 4 LDS P-loads + 16 V-operand pairs before the 16 PV WMMAs.
// ---------------------------------------------------------------------------
__global__ __launch_bounds__(128) void attn_kernel(
    const float* __restrict__ qkv, float* __restrict__ attn) {
  __shared__ float Plds[4 * 16 * 18];  // 18-float row pitch keeps b64 aligned
  const int lane = threadIdx.x & 31;
  const int wave = threadIdx.x >> 5;
  const int lo = lane & 15;
  const int hi = lane >> 4;

  const int blk = blockIdx.x;
  const int qgrp = blk % (T_SEQ / 64);
  const int bh = blk / (T_SEQ / 64);
  const int h = bh % NHEAD;
  const int b = bh / NHEAD;
  const int qb = qgrp * 64 + wave * 16;

  const unsigned rs = 3 * D_MODEL;  // qkv row stride (elements)
  const float* base = qkv + (size_t)b * T_SEQ * rs;
  const float* kbase = base + D_MODEL + h * HDIM;
  const float* vbase = base + 2 * D_MODEL + h * HDIM;

  // Preload Q tile in A-operand layout: lane -> row qb+lo, k = 4*i + 2*hi.
  v2f qa[16];
  {
    const float* qrow = base + (size_t)(qb + lo) * rs + h * HDIM + 2 * hi;
#pragma unroll
    for (int i = 0; i < 16; ++i) qa[i] = *(const v2f*)(qrow + 4 * i);
  }

  v8f acc[4];
#pragma unroll
  for (int nt = 0; nt < 4; ++nt) acc[nt] = (v8f)(0.0f);
  float mrow[8], lrow[8];
#pragma unroll
  for (int j = 0; j < 8; ++j) { mrow[j] = -1.0e30f; lrow[j] = 0.0f; }

  float* myP = Plds + wave * (16 * 18);
  const float scale = 0.125f;  // hd^-0.5, hd=64

  for (int kb = 0; kb <= qb; kb += 16) {
    // ---- batch-load K-tile operands, then S = Q @ K^T
    const float* krow = kbase + (size_t)(kb + lo) * rs + 2 * hi;
    v2f kv[16];
#pragma unroll
    for (int i = 0; i < 16; ++i) kv[i] = *(const v2f*)(krow + 4 * i);
    v8f s = (v8f)(0.0f);
#pragma unroll
    for (int i = 0; i < 16; ++i) s = wmma4(qa[i], kv[i], s);

    // ---- online softmax per row (row m = j + 8*hi, col n = lo)
    const bool diag = (kb == qb);
#pragma unroll
    for (int j = 0; j < 8; ++j) {
      float sv = s[j] * scale;
      if (diag) {
        const int m = j + 8 * hi;
        if (lo > m) sv = -1.0e30f;  // causal mask within diagonal tile
      }
      float tmax = sv;
      tmax = fmaxf(tmax, __shfl_xor(tmax, 1));
      tmax = fmaxf(tmax, __shfl_xor(tmax, 2));
      tmax = fmaxf(tmax, __shfl_xor(tmax, 4));
      tmax = fmaxf(tmax, __shfl_xor(tmax, 8));
      const float mnew = fmaxf(mrow[j], tmax);
      const float al = __expf(mrow[j] - mnew);
      const float p = __expf(sv - mnew);
      float rsum = p;
      rsum += __shfl_xor(rsum, 1);
      rsum += __shfl_xor(rsum, 2);
      rsum += __shfl_xor(rsum, 4);
      rsum += __shfl_xor(rsum, 8);
      lrow[j] = lrow[j] * al + rsum;
      mrow[j] = mnew;
#pragma unroll
      for (int nt = 0; nt < 4; ++nt) acc[nt][j] *= al;
      myP[(j + 8 * hi) * 18 + lo] = p;  // stage P for A-layout reload
    }

    // ---- batch-load P (A-layout from LDS) and V operands, then O += P @ V
    v2f pa[4];
#pragma unroll
    for (int i = 0; i < 4; ++i)
      pa[i] = *(const v2f*)(myP + lo * 18 + 4 * i + 2 * hi);
    v2f vb[4][4];
#pragma unroll
    for (int i = 0; i < 4; ++i) {
      const float* vr0 = vbase + (size_t)(kb + 4 * i + 2 * hi) * rs + lo;
#pragma unroll
      for (int nt = 0; nt < 4; ++nt) {
        v2f tt;
        tt.x = vr0[nt * 16];
        tt.y = vr0[rs + nt * 16];
        vb[i][nt] = tt;
      }
    }
#pragma unroll
    for (int i = 0; i < 4; ++i)
#pragma unroll
      for (int nt = 0; nt < 4; ++nt)
        acc[nt] = wmma4(pa[i], vb[i][nt], acc[nt]);
  }

  // ---- epilogue: O /= l, scatter to attn[b, qb+m, h*64 + nt*16 + n]
  float inv[8];
#pragma unroll
  for (int j = 0; j < 8; ++j) inv[j] = 1.0f / lrow[j];
  float* obase = attn + ((size_t)b * T_SEQ + qb) * D_MODEL + h * HDIM;
#pragma unroll
  for (int nt = 0; nt < 4; ++nt)
#pragma unroll
    for (int j = 0; j < 8; ++j) {
      const int m = j + 8 * hi;
      obase[(size_t)m * D_MODEL + nt * 16 + lo] = acc[nt][j] * inv[j];
    }
}

// ---------------------------------------------------------------------------
// attn[b,t,:] += bias_scale * sum_k emb[seq[b,t,k], :]   (bandwidth-bound)
// ---------------------------------------------------------------------------
__global__ __launch_bounds__(256) void bias_kernel(
    float* __restrict__ attn, const int* __restrict__ seq,
    const float* __restrict__ emb, const float* __restrict__ bscale) {
  const int bt = blockIdx.x;
  const int d0 = threadIdx.x * 4;
  const float bs = bscale[0];
  float s0 = 0.f, s1 = 0.f, s2 = 0.f, s3 = 0.f;
#pragma unroll
  for (int k = 0; k < KTPL; ++k) {
    const int idx = seq[bt * KTPL + k];
    const float4 ev = *(const float4*)(emb + (size_t)idx * D_MODEL + d0);
    s0 += ev.x; s1 += ev.y; s2 += ev.z; s3 += ev.w;
  }
  float* o = attn + (size_t)bt * D_MODEL + d0;
  float4 ov = *(const float4*)o;
  ov.x += bs * s0; ov.y += bs * s1; ov.z += bs * s2; ov.w += bs * s3;
  *(float4*)o = ov;
}

// ---------------------------------------------------------------------------
extern "C" void kernel_launch(void* const* d_in, const int* in_sizes, int n_in,
                              void* d_out, int out_size, void* d_ws,
                              size_t ws_size, hipStream_t stream) {
  (void)in_sizes; (void)n_in; (void)out_size; (void)ws_size;
  const float* x      = (const float*)d_in[0];
  const int*   seq    = (const int*)d_in[1];
  const float* Wqkv   = (const float*)d_in[2];
  const float* Wout   = (const float*)d_in[3];
  const float* bout   = (const float*)d_in[4];
  const float* emb    = (const float*)d_in[5];
  const float* bscale = (const float*)d_in[6];
  float* out = (float*)d_out;

  float* qkv  = (float*)d_ws;                                  // [B,T,3D]
  float* attn = qkv + (size_t)BATCH * T_SEQ * 3 * D_MODEL;     // [B,T,D]

  const int M = BATCH * T_SEQ;  // 8192

  // 1) qkv = x @ Wqkv^T        [8192,1024] x [3072,1024]^T
  gemm_nt_kernel<<<dim3((3 * D_MODEL) / 256, M / 64), dim3(256), 0, stream>>>(
      x, Wqkv, nullptr, qkv, M, 3 * D_MODEL, D_MODEL);

  // 2) causal flash attention -> attn [B,T,D]
  attn_kernel<<<dim3(BATCH * NHEAD * (T_SEQ / 64)), dim3(128), 0, stream>>>(
      qkv, attn);

  // 3) attn += bias_scale * sum(emb[sequences], axis=-2)
  bias_kernel<<<dim3(BATCH * T_SEQ), dim3(256), 0, stream>>>(attn, seq, emb,
                                                             bscale);

  // 4) out = attn @ Wout^T + bout
  gemm_nt_kernel<<<dim3(D_MODEL / 256, M / 64), dim3(256), 0, stream>>>(
      attn, Wout, bout, out, M, D_MODEL, D_MODEL);
}